// VDP_Maxpool_26499948216841
// MI455X (gfx1250) — compile-verified
//
#include <hip/hip_runtime.h>
#include <stdint.h>

// Problem constants (from reference): B=8, C=16, H=W=32 -> N=1024, P=256
#define BC      128          // B*C
#define NFLAT   1024         // H*W
#define POOLN   256          // P = 16*16
#define MU_OUT  (BC * POOLN) // 32768 floats of mu_p at head of d_out

// ---------------------------------------------------------------------------
// Kernel 1: 2x2/stride-2 maxpool with flat argmax index (first-max tie-break,
// matching jnp.argmax). One thread per pooled element (32768 total).
// ---------------------------------------------------------------------------
__global__ void vdp_pool_kernel(const float* __restrict__ mu,
                                float* __restrict__ mu_p,
                                int* __restrict__ idx) {
  int t  = blockIdx.x * blockDim.x + threadIdx.x;   // 0..32767
  int w  = t & 15;
  int h  = (t >> 4) & 15;
  int bc = t >> 8;
  const float* base = mu + (size_t)bc * NFLAT + (h * 2) * 32 + (w * 2);
  float v00 = base[0];
  float v01 = base[1];
  float v10 = base[32];
  float v11 = base[33];
  // window order (r0c0, r0c1, r1c0, r1c1); strict '>' keeps first max on ties
  float m = v00; int k = 0;
  if (v01 > m) { m = v01; k = 1; }
  if (v10 > m) { m = v10; k = 2; }
  if (v11 > m) { m = v11; k = 3; }
  mu_p[t] = m;
  idx[t]  = (h * 2 + (k >> 1)) * 32 + (w * 2 + (k & 1));  // flat over H*W
}

// ---------------------------------------------------------------------------
// CDNA5 async-tensor helpers (gfx1250): GLOBAL_LOAD_ASYNC_TO_LDS_B128 copies
// 16B/lane (512B/wave-op) from global memory into LDS, tracked with ASYNCcnt.
// ---------------------------------------------------------------------------
__device__ __forceinline__ unsigned lds_addr_of(const void* p) {
  // Generic shared address = {aperture_hi, lds_offset}; the low 32 bits are
  // exactly the LDS byte offset used by DS / async-LDS addressing.
  return (unsigned)(unsigned long long)p;
}

template <int N>
__device__ __forceinline__ void wait_asynccnt_le() {
  asm volatile("s_wait_asynccnt %0" ::"n"(N) : "memory");
}

// Stage one 4KB sigma row (1024 f32) into LDS: 8 async B128 ops per wave.
// INST_OFFSET is applied to BOTH the global address and the LDS address, so a
// single (lds, global) VGPR setup covers the whole row.
__device__ __forceinline__ void async_row_load(unsigned ldsBase,
                                               const float* rowPtr,
                                               int lane) {
  unsigned    l = ldsBase + (unsigned)(lane * 16);
  const char* g = (const char*)rowPtr + lane * 16;
  asm volatile(
      "s_wait_dscnt 0x0\n\t"  // close DS-vs-async WAR window on this buffer
      "global_load_async_to_lds_b128 %0, %1, off\n\t"
      "global_load_async_to_lds_b128 %0, %1, off offset:512\n\t"
      "global_load_async_to_lds_b128 %0, %1, off offset:1024\n\t"
      "global_load_async_to_lds_b128 %0, %1, off offset:1536\n\t"
      "global_load_async_to_lds_b128 %0, %1, off offset:2048\n\t"
      "global_load_async_to_lds_b128 %0, %1, off offset:2560\n\t"
      "global_load_async_to_lds_b128 %0, %1, off offset:3072\n\t"
      "global_load_async_to_lds_b128 %0, %1, off offset:3584"
      :
      : "v"(l), "v"(g)
      : "memory");
}

// ---------------------------------------------------------------------------
// Kernel 2: sigma_p[bc, i, j] = sigma[bc, idx[i], idx[j]]
// grid = (128 bc, 8 row-chunks), block = 256 threads = 8 wave32s.
// Each wave owns 2 private 4KB LDS row buffers (double-buffered, no barriers).
// Each lane caches its 8 column indices in registers for the whole block.
// ---------------------------------------------------------------------------
__global__ void __launch_bounds__(256)
vdp_gather_kernel(const float* __restrict__ sigma,
                  const int* __restrict__ idx,
                  float* __restrict__ sigma_p) {
  __shared__ float rowbuf[8][2][NFLAT];  // 8 waves x 2 buffers x 4KB = 64KB

  const int bc    = blockIdx.x;        // 0..127
  const int chunk = blockIdx.y;        // 0..7 -> rows [chunk*32, chunk*32+32)
  const int lane  = threadIdx.x & 31;
  const int wave  = threadIdx.x >> 5;  // 0..7

  const int*   myidx = idx + bc * POOLN;
  const float* sbase = sigma + (size_t)bc * NFLAT * NFLAT;
  float*       obase = sigma_p + (size_t)bc * POOLN * POOLN;

  // Per-lane column indices j = j0*32 + lane, j0 = 0..7 (reused for this wave's 4 rows).
  int joff[8];
#pragma unroll
  for (int j0 = 0; j0 < 8; ++j0) joff[j0] = myidx[j0 * 32 + lane];

  // This wave's 4 output rows: i = chunk*32 + wave + 8*k
  const int i0 = chunk * 32 + wave;
  int ridx[4];
#pragma unroll
  for (int k = 0; k < 4; ++k) ridx[k] = myidx[i0 + 8 * k];

  const unsigned lds0 = lds_addr_of(&rowbuf[wave][0][0]);
  const unsigned lds1 = lds_addr_of(&rowbuf[wave][1][0]);

  // Prime the pipeline: row 0 -> buffer 0.
  async_row_load(lds0, sbase + (size_t)ridx[0] * NFLAT, lane);

#pragma unroll
  for (int k = 0; k < 4; ++k) {
    if (k < 3) {
      // Prefetch row k+1 into the other buffer ((k+1)&1).
      async_row_load((k & 1) ? lds0 : lds1,
                     sbase + (size_t)ridx[k + 1] * NFLAT, lane);
      wait_asynccnt_le<8>();   // the 8 oldest (row k) ops are complete
    } else {
      wait_asynccnt_le<0>();   // last row: drain everything
    }
    // Column-gather row k from LDS, store coalesced (32 lanes x b32).
    float* orow = obase + (size_t)(i0 + 8 * k) * POOLN;
#pragma unroll
    for (int j0 = 0; j0 < 8; ++j0) {
      orow[j0 * 32 + lane] = rowbuf[wave][k & 1][joff[j0]];
    }
  }
}

// ---------------------------------------------------------------------------
// Launch: d_out = [mu_p (32768 f32) | sigma_p (8388608 f32)], d_ws holds the
// 32768 int argmax indices (128KB).
// ---------------------------------------------------------------------------
extern "C" void kernel_launch(void* const* d_in, const int* in_sizes, int n_in,
                              void* d_out, int out_size, void* d_ws,
                              size_t ws_size, hipStream_t stream) {
  const float* mu    = (const float*)d_in[0];
  const float* sigma = (const float*)d_in[1];
  float* out     = (float*)d_out;
  float* mu_p    = out;
  float* sigma_p = out + MU_OUT;
  int*   idx     = (int*)d_ws;

  vdp_pool_kernel<<<dim3(BC * POOLN / 256), dim3(256), 0, stream>>>(mu, mu_p,
                                                                    idx);
  vdp_gather_kernel<<<dim3(BC, 8), dim3(256), 0, stream>>>(sigma, idx,
                                                           sigma_p);
}